// TriplaneEncoding_90220083020078
// MI455X (gfx1250) — compile-verified
//
#include <hip/hip_runtime.h>
#include <stdint.h>

// Problem constants (match reference: RES, LEVEL, C, B)
#define RES_   512
#define LEVEL_ 8
#define C_     32
#define B_     524288
#define TOT_   349520   // sum_{l=0..7} (512>>l)^2 pixels per (plane,channel)

static const int h_offs[LEVEL_] = {
    0, 262144, 327680, 344064, 348160, 349184, 349440, 349504};

typedef unsigned int u32x4 __attribute__((ext_vector_type(4)));
typedef int          i32x8 __attribute__((ext_vector_type(8)));
typedef int          i32x4 __attribute__((ext_vector_type(4)));
typedef float        f32x4 __attribute__((ext_vector_type(4)));

// closed form for mip offset: off(l) = (2^20 - 2^(20-2l)) / 3
__device__ __forceinline__ uint32_t mip_off(int l) {
    return (0x100000u - (0x100000u >> (2 * l))) / 3u;
}

// ---------------------------------------------------------------------------
// K1: transpose level 0 of mat (plane, chan, pix) -> pool (plane, pix, chan).
// One wave per 4 pixels: each lane loads a float4 of its channel (b128,
// non-temporal: mat is never read again), wave emits 4 coalesced 128B stores.
// ---------------------------------------------------------------------------
__global__ void k_transpose(const float* __restrict__ mat,
                            float* __restrict__ pool) {
    int wave = (int)((blockIdx.x * blockDim.x + threadIdx.x) >> 5);
    int lane = threadIdx.x & 31;
    int grp  = wave % (RES_ * RES_ / 4);       // pixel-quad within plane
    int p    = wave / (RES_ * RES_ / 4);
    int pix0 = grp * 4;

    const f32x4* src = (const f32x4*)(mat +
        ((size_t)p * C_ + lane) * (RES_ * RES_) + pix0);
    f32x4 v = __builtin_nontemporal_load(src);

    float* dst = pool + ((size_t)p * TOT_ + pix0) * C_ + lane;
    dst[0 * C_] = v.x;
    dst[1 * C_] = v.y;
    dst[2 * C_] = v.z;
    dst[3 * C_] = v.w;
}

// ---------------------------------------------------------------------------
// K2: 2x2 average-pool level (l-1) -> level l, channel-last both sides.
// The two input rows for a block are staged into LDS with the Tensor Data
// Mover (tensor_load_to_lds, 2D descriptor), synchronized via TENSORcnt.
// ---------------------------------------------------------------------------
__global__ void k_downsample(float* __restrict__ pool,
                             int res_in, int res_out,
                             int off_in, int off_out, int chunk) {
    // max tile: 2 rows x (64 out px * 2 in px * 32 ch) floats = 32 KB
    __shared__ float smem[2 * 64 * 2 * C_];

    int bpr       = res_out / chunk;       // x-chunks per row
    int per_plane = res_out * bpr;
    int p   = blockIdx.x / per_plane;
    int rem = blockIdx.x % per_plane;
    int yo  = rem / bpr;                   // output row
    int xc  = rem % bpr;                   // x-chunk index
    int tile_elems = chunk * 2 * C_;       // elements per input row of tile

    const float* src = pool + ((size_t)p * TOT_ + off_in +
                               (size_t)(2 * yo) * res_in +
                               (size_t)xc * chunk * 2) * C_;

    if ((threadIdx.x >> 5) == 0) {
        uint64_t ga = (uint64_t)(uintptr_t)src;          // global byte address
        // low 32 bits of a flat LDS pointer == LDS byte offset on gfx1250
        uint32_t lds_addr = (uint32_t)(uintptr_t)(&smem[0]);

        // --- D# group 0: count=1, lds_addr[63:32], global_addr[120:64], type=2
        u32x4 g0;
        g0[0] = 1u;                                       // count = 1 descriptor
        g0[1] = lds_addr;
        g0[2] = (uint32_t)ga;
        g0[3] = ((uint32_t)(ga >> 32) & 0x1FFFFFFu) | (2u << 30);

        // --- D# group 1: data_size=4B, dims/strides of 2D tile (2 rows)
        uint64_t stride0 = (uint64_t)res_in * C_;        // elems between rows
        uint32_t td0 = (uint32_t)tile_elems;             // tensor_dim0 (no OOB)
        uint32_t td1 = 2u;                               // tensor_dim1 = 2 rows
        i32x8 g1;
        g1[0] = (int)(2u << 16);                         // data_size = 2 (4B)
        g1[1] = (int)((td0 & 0xFFFFu) << 16);            // tensor_dim0[15:0]
        g1[2] = (int)(((td0 >> 16) & 0xFFFFu) |          // tensor_dim0[31:16]
                      ((td1 & 0xFFFFu) << 16));          // tensor_dim1[15:0]
        g1[3] = (int)(((td1 >> 16) & 0xFFFFu) |          // tensor_dim1[31:16]
                      ((td0 & 0xFFFFu) << 16));          // tile_dim0 = tile_elems
        g1[4] = (int)2;                                  // tile_dim1=2, tile_dim2=0
        g1[5] = (int)(uint32_t)stride0;                  // tensor_dim0_stride lo
        g1[6] = (int)((uint32_t)(stride0 >> 32) & 0xFFFFu); // stride hi, dim1_stride=0
        g1[7] = 0;
        i32x4 gz = {0, 0, 0, 0};                         // groups 2/3: unused dims

        asm volatile("tensor_load_to_lds %0, %1, %2, %3"
                     :: "s"(g0), "s"(g1), "s"(gz), "s"(gz)
                     : "memory");
        __builtin_amdgcn_s_wait_tensorcnt(0);
    }
    __syncthreads();

    int lane = threadIdx.x & 31;
    int wv   = threadIdx.x >> 5;
    float* dst = pool + ((size_t)p * TOT_ + off_out +
                         (size_t)yo * res_out + (size_t)xc * chunk) * C_;
    for (int xo = wv; xo < chunk; xo += 8) {
        float v00 = smem[(2 * xo)     * C_ + lane];
        float v01 = smem[(2 * xo + 1) * C_ + lane];
        float v10 = smem[tile_elems + (2 * xo)     * C_ + lane];
        float v11 = smem[tile_elems + (2 * xo + 1) * C_ + lane];
        dst[(size_t)xo * C_ + lane] = 0.25f * (v00 + v01 + v10 + v11);
    }
}

// ---------------------------------------------------------------------------
// K3: trilinear gather. One wave per point, lane = channel.
// All texel addresses are 32-bit byte offsets off the uniform pool base
// (pool is 134 MB < 4 GiB) -> global_load saddr form, minimal VALU.
// 24 independent coalesced 128B loads per point -> deep memory parallelism.
// Output stores are non-temporal so the 201 MB output stream does not evict
// the L2-resident pool.
// ---------------------------------------------------------------------------
__global__ void k_gather(const float* __restrict__ x,
                         const float* __restrict__ r,
                         const float* __restrict__ pool,
                         float* __restrict__ out) {
    int b    = (int)((blockIdx.x * blockDim.x + threadIdx.x) >> 5);  // point
    int lane = threadIdx.x & 31;                                     // channel

    float x0 = x[3 * (size_t)b + 0];
    float x1 = x[3 * (size_t)b + 1];
    float x2 = x[3 * (size_t)b + 2];
    float rv = r[b];

    float lvl = fminf(fmaxf(log2f(1024.0f * rv), 0.0f), (float)(LEVEL_ - 1));
    int   l0  = (int)floorf(lvl);
    int   l1  = min(l0 + 1, LEVEL_ - 1);
    float w   = lvl - (float)l0;

    const float ux[3] = {x0, x1, x2};   // perms: (0,1), (1,2), (2,0)
    const float uy[3] = {x1, x2, x0};

    float acc[3] = {0.0f, 0.0f, 0.0f};
    const char* pc = (const char*)pool;
    uint32_t lane_b = (uint32_t)lane * 4u;

#pragma unroll
    for (int t = 0; t < 2; ++t) {
        int   l  = (t == 0) ? l0 : l1;
        float wt = (t == 0) ? (1.0f - w) : w;
        int   res_i = RES_ >> l;
        float scale = (float)(res_i - 1);
        uint32_t lvl_b = mip_off(l) * (C_ * 4u);         // level base (bytes)

#pragma unroll
        for (int p = 0; p < 3; ++p) {
            float u = ux[p] * scale;
            float v = uy[p] * scale;
            int ix0 = min(max((int)floorf(u), 0), res_i - 1);
            int iy0 = min(max((int)floorf(v), 0), res_i - 1);
            int ix1 = min(ix0 + 1, res_i - 1);
            int iy1 = min(iy0 + 1, res_i - 1);
            float fx = fminf(fmaxf(u - (float)ix0, 0.0f), 1.0f);
            float fy = fminf(fmaxf(v - (float)iy0, 0.0f), 1.0f);

            uint32_t base = (uint32_t)p * (TOT_ * C_ * 4u) + lvl_b + lane_b;
            uint32_t r0 = (uint32_t)(iy0 * res_i) * (C_ * 4u);
            uint32_t r1 = (uint32_t)(iy1 * res_i) * (C_ * 4u);
            uint32_t c0 = (uint32_t)ix0 * (C_ * 4u);
            uint32_t c1 = (uint32_t)ix1 * (C_ * 4u);

            float v00 = *(const float*)(pc + (base + r0 + c0));
            float v10 = *(const float*)(pc + (base + r0 + c1));
            float v01 = *(const float*)(pc + (base + r1 + c0));
            float v11 = *(const float*)(pc + (base + r1 + c1));

            float bil = v00 * (1.0f - fx) * (1.0f - fy) +
                        v10 * fx * (1.0f - fy) +
                        v01 * (1.0f - fx) * fy +
                        v11 * fx * fy;
            acc[p] += wt * bil;
        }
    }

    float* o = out + (size_t)b * (3 * C_) + lane;
    __builtin_nontemporal_store(acc[0], o + 0 * C_);
    __builtin_nontemporal_store(acc[1], o + 1 * C_);
    __builtin_nontemporal_store(acc[2], o + 2 * C_);
}

// ---------------------------------------------------------------------------
extern "C" void kernel_launch(void* const* d_in, const int* in_sizes, int n_in,
                              void* d_out, int out_size, void* d_ws, size_t ws_size,
                              hipStream_t stream) {
    const float* x   = (const float*)d_in[0];   // (B,3)
    const float* r   = (const float*)d_in[1];   // (B,)
    const float* mat = (const float*)d_in[2];   // (3,C,RES,RES)
    float* out  = (float*)d_out;                // (B, 96)
    float* pool = (float*)d_ws;                 // 3*TOT_*C_ floats = 128 MiB

    // K1: transpose level 0 into channel-last pool (wave per pixel-quad)
    {
        int waves  = 3 * (RES_ * RES_ / 4);
        int blocks = waves / 8;                 // 256 threads = 8 waves
        k_transpose<<<blocks, 256, 0, stream>>>(mat, pool);
    }

    // K2: build mip levels 1..7 (chained; TDM-staged tiles)
    int res_in = RES_;
    for (int l = 1; l < LEVEL_; ++l) {
        int res_out = RES_ >> l;
        int chunk   = res_out < 64 ? res_out : 64;   // out pixels per block
        int bpr     = res_out / chunk;
        int blocks  = 3 * res_out * bpr;
        k_downsample<<<blocks, 256, 0, stream>>>(pool, res_in, res_out,
                                                 h_offs[l - 1], h_offs[l], chunk);
        res_in = res_out;
    }

    // K3: gather (one wave per point)
    {
        int blocks = B_ / 8;                    // 8 points per 256-thread block
        k_gather<<<blocks, 256, 0, stream>>>(x, r, pool, out);
    }
}